// GINNet_35399120454208
// MI455X (gfx1250) — compile-verified
//
#include <hip/hip_runtime.h>
#include <hip/hip_bf16.h>

typedef __attribute__((ext_vector_type(2))) float v2f;
typedef __attribute__((ext_vector_type(8))) float v8f;

#define N_NODES  50000
#define N_EDGES  800000
#define N_GRAPHS 1000
#define G_PAD    1008      // 1000 padded up to a multiple of 16
#define BN_EPS   1e-5f

#if defined(__HIP_DEVICE_COMPILE__)
#if !__has_builtin(__builtin_amdgcn_wmma_f32_16x16x4_f32)
#error "DEVICE PASS: wmma_f32_16x16x4_f32 builtin not available"
#endif
#endif

// ---------------------------------------------------------------- utilities
__global__ void zero_f32_kernel(float* __restrict__ p, int n) {
  int i = blockIdx.x * blockDim.x + threadIdx.x;
  if (i < n) p[i] = 0.0f;
}

__global__ void copy_vec4_kernel(float4* __restrict__ dst, const float4* __restrict__ src, int n4) {
  int i = blockIdx.x * blockDim.x + threadIdx.x;
  if (i < n4) dst[i] = src[i];
}

// Copy [R,11] -> [R,12] with zero pad column (rows beyond 11 input cols -> 0).
__global__ void pad12_kernel(const float* __restrict__ x, float* __restrict__ out, int n12) {
  int i = blockIdx.x * blockDim.x + threadIdx.x;
  if (i >= n12) return;
  int r = i / 12;
  int f = i - r * 12;
  out[i] = (f < 11) ? x[r * 11 + f] : 0.0f;
}

// ------------------------------------------------- GIN edge scatter (atomic)
__global__ void edge_scatter_pad(const float* __restrict__ h,
                                 const int* __restrict__ srcIdx,
                                 const int* __restrict__ dstIdx,
                                 float* __restrict__ out, int E, int F, int outStride) {
  int tid = blockIdx.x * blockDim.x + threadIdx.x;
  if (tid >= E * F) return;
  int e = tid / F;
  int f = tid - e * F;
  atomicAdd(&out[(size_t)dstIdx[e] * outStride + f], h[(size_t)srcIdx[e] * F + f]);
}

__global__ void edge_scatter_vec4(const float4* __restrict__ h,
                                  const int* __restrict__ srcIdx,
                                  const int* __restrict__ dstIdx,
                                  float* __restrict__ out, int E, int F4) {
  int tid = blockIdx.x * blockDim.x + threadIdx.x;
  if (tid >= E * F4) return;
  int e  = tid / F4;
  int f4 = tid - e * F4;
  float4 v = h[(size_t)srcIdx[e] * F4 + f4];
  float* o = out + ((size_t)dstIdx[e] * F4 + f4) * 4;
  atomicAdd(o + 0, v.x);
  atomicAdd(o + 1, v.y);
  atomicAdd(o + 2, v.z);
  atomicAdd(o + 3, v.w);
}

// ------------------------------------------------------------ graph pooling
__global__ void pool_scatter_vec4(const float4* __restrict__ h,
                                  const int* __restrict__ batch,
                                  float* __restrict__ out, int Nn, int F4) {
  int tid = blockIdx.x * blockDim.x + threadIdx.x;
  if (tid >= Nn * F4) return;
  int n  = tid / F4;
  int f4 = tid - n * F4;
  float4 v = h[(size_t)n * F4 + f4];
  float* o = out + ((size_t)batch[n] * F4 + f4) * 4;
  atomicAdd(o + 0, v.x);
  atomicAdd(o + 1, v.y);
  atomicAdd(o + 2, v.z);
  atomicAdd(o + 3, v.w);
}

// ---------------------------------------------------- WMMA f32 GEMM + bias
// Y[M,N] = A[M,K] @ W[K,N] + bias[N].  Requires M%16==0, K%4==0, N%64==0.
// One wave -> 16x64 strip: one A fragment feeds 4 WMMAs per K-step.
__global__ void gemm_wmma_f32_fast(const float* __restrict__ A, const float* __restrict__ W,
                                   const float* __restrict__ bias, float* __restrict__ Y,
                                   int M, int K, int N) {
  const int lane = threadIdx.x & 31;
  const int wave = blockIdx.x * (blockDim.x >> 5) + (threadIdx.x >> 5);
  const int stripsN = N >> 6;
  const int tilesM  = M >> 4;
  if (wave >= tilesM * stripsN) return;  // wave-uniform: EXEC stays all-1

  const int tm   = wave / stripsN;
  const int tn   = wave - tm * stripsN;
  const int row0 = tm << 4;
  const int col0 = tn << 6;
  const int lm   = lane & 15;
  const int lh   = lane >> 4;

  const float* __restrict__ Arow = A + (size_t)(row0 + lm) * K + lh * 2;
  const float* __restrict__ Wp   = W + (size_t)(lh * 2) * N + col0 + lm;

  v8f acc0 = {}, acc1 = {}, acc2 = {}, acc3 = {};
  for (int k = 0; k < K; k += 4) {
    const v2f a = *(const v2f*)(Arow + k);
    const float* __restrict__ wk = Wp + (size_t)k * N;
    v2f b0, b1, b2, b3;
    b0.x = wk[0];      b0.y = wk[N];
    b1.x = wk[16];     b1.y = wk[N + 16];
    b2.x = wk[32];     b2.y = wk[N + 32];
    b3.x = wk[48];     b3.y = wk[N + 48];
#if defined(__HIP_DEVICE_COMPILE__)
    acc0 = __builtin_amdgcn_wmma_f32_16x16x4_f32(false, a, false, b0, (short)0, acc0, false, false);
    acc1 = __builtin_amdgcn_wmma_f32_16x16x4_f32(false, a, false, b1, (short)0, acc1, false, false);
    acc2 = __builtin_amdgcn_wmma_f32_16x16x4_f32(false, a, false, b2, (short)0, acc2, false, false);
    acc3 = __builtin_amdgcn_wmma_f32_16x16x4_f32(false, a, false, b3, (short)0, acc3, false, false);
#endif
  }

  const float bc0 = bias[col0 + lm];
  const float bc1 = bias[col0 + 16 + lm];
  const float bc2 = bias[col0 + 32 + lm];
  const float bc3 = bias[col0 + 48 + lm];
#pragma unroll
  for (int i = 0; i < 8; ++i) {
    float* yr = Y + (size_t)(row0 + i + lh * 8) * N + col0 + lm;
    yr[0]  = acc0[i] + bc0;
    yr[16] = acc1[i] + bc1;
    yr[32] = acc2[i] + bc2;
    yr[48] = acc3[i] + bc3;
  }
}

// -------------------------------------------------------- BatchNorm (train)
__global__ void bn_stats_kernel(const float* __restrict__ Y,
                                float* __restrict__ sums, float* __restrict__ sumsq,
                                int Rows, int F) {
  int col = threadIdx.x;
  if (col >= F) return;
  float s = 0.0f, q = 0.0f;
  for (int r = blockIdx.x; r < Rows; r += gridDim.x) {
    float v = Y[(size_t)r * F + col];
    s += v;
    q += v * v;
  }
  atomicAdd(&sums[col], s);
  atomicAdd(&sumsq[col], q);
}

__global__ void bn_finalize_kernel(const float* __restrict__ sums,
                                   const float* __restrict__ sumsq,
                                   const float* __restrict__ g,
                                   const float* __restrict__ be,
                                   float* __restrict__ scale, float* __restrict__ shift,
                                   int Rows, int F) {
  int c = blockIdx.x * blockDim.x + threadIdx.x;
  if (c >= F) return;
  float inv = 1.0f / (float)Rows;
  float m = sums[c] * inv;
  float v = sumsq[c] * inv - m * m;
  float rstd = rsqrtf(v + BN_EPS);
  float sc = g[c] * rstd;
  scale[c] = sc;
  shift[c] = be[c] - m * sc;
}

__global__ void bn_apply_relu_kernel(float* __restrict__ Y,
                                     const float* __restrict__ scale,
                                     const float* __restrict__ shift,
                                     int total, int F) {
  int i = blockIdx.x * blockDim.x + threadIdx.x;
  if (i >= total) return;
  int c = i % F;
  float v = Y[i] * scale[c] + shift[c];
  Y[i] = v > 0.0f ? v : 0.0f;
}

// -------------------------------------------------- final [G,K] @ [K,1] + b
__global__ void final_linear_kernel(const float* __restrict__ X,
                                    const float* __restrict__ w,
                                    const float* __restrict__ b,
                                    float* __restrict__ out, int K) {
  __shared__ float red[128];
  int row = blockIdx.x;
  int t = threadIdx.x;
  float s = 0.0f;
  for (int k = t; k < K; k += blockDim.x) s += X[(size_t)row * K + k] * w[k];
  red[t] = s;
  __syncthreads();
  for (int off = blockDim.x >> 1; off > 0; off >>= 1) {
    if (t < off) red[t] += red[t + off];
    __syncthreads();
  }
  if (t == 0) out[row] = red[0] + b[0];
}

// ---------------------------------------------------------------- dispatch
extern "C" void kernel_launch(void* const* d_in, const int* in_sizes, int n_in,
                              void* d_out, int out_size, void* d_ws, size_t ws_size,
                              hipStream_t stream) {
  (void)in_sizes; (void)n_in; (void)out_size; (void)ws_size;

  const float* x      = (const float*)d_in[0];
  const int*   ei     = (const int*)d_in[1];
  const int*   srcIdx = ei;                 // edge_index[0]
  const int*   dstIdx = ei + N_EDGES;       // edge_index[1]
  const int*   batch  = (const int*)d_in[2];

  const float *Wm[6], *Bm[6], *Gm[6], *BEm[6];
  for (int l = 0; l < 6; ++l) {
    Wm[l]  = (const float*)d_in[3 + 4 * l + 0];
    Bm[l]  = (const float*)d_in[3 + 4 * l + 1];
    Gm[l]  = (const float*)d_in[3 + 4 * l + 2];
    BEm[l] = (const float*)d_in[3 + 4 * l + 3];
  }
  const float* fw1 = (const float*)d_in[27];
  const float* fb1 = (const float*)d_in[28];
  const float* fg  = (const float*)d_in[29];
  const float* fbe = (const float*)d_in[30];
  const float* fw2 = (const float*)d_in[31];
  const float* fb2 = (const float*)d_in[32];
  float* out = (float*)d_out;

  // workspace layout
  float* B0    = (float*)d_ws;                          // [50000,256]
  float* B1    = B0 + (size_t)N_NODES * 256;            // [50000,256]
  float* hg    = B1 + (size_t)N_NODES * 256;            // [1008,256] padded
  float* hgl   = hg + (size_t)G_PAD * 256;              // [1008,128] padded
  float* W0p   = hgl + (size_t)G_PAD * 128;             // [12,128] padded l1w
  float* stats = W0p + 12 * 128;                        // 1024 floats
  float* sums  = stats;
  float* sumsq = stats + 256;
  float* scale = stats + 512;
  float* shift = stats + 768;

  auto cdiv = [](int a, int b) { return (a + b - 1) / b; };

  auto gemm = [&](const float* A, const float* W, const float* bias, float* Y,
                  int M, int K, int N) {
    int tiles  = (M / 16) * (N / 64);
    int blocks = cdiv(tiles, 4);
    gemm_wmma_f32_fast<<<blocks, 128, 0, stream>>>(A, W, bias, Y, M, K, N);
  };

  auto bn_relu = [&](float* Y, const float* g, const float* be, int Rows, int F) {
    zero_f32_kernel<<<2, 256, 0, stream>>>(stats, 512);
    bn_stats_kernel<<<256, F, 0, stream>>>(Y, sums, sumsq, Rows, F);
    bn_finalize_kernel<<<1, 256, 0, stream>>>(sums, sumsq, g, be, scale, shift, Rows, F);
    int total = Rows * F;
    bn_apply_relu_kernel<<<cdiv(total, 256), 256, 0, stream>>>(Y, scale, shift, total, F);
  };

  auto gin_agg = [&](const float* h, float* dst, int F) {
    int n = N_NODES * F;
    copy_vec4_kernel<<<cdiv(n / 4, 256), 256, 0, stream>>>((float4*)dst, (const float4*)h, n / 4);
    int F4 = F >> 2;
    edge_scatter_vec4<<<cdiv(N_EDGES * F4, 256), 256, 0, stream>>>(
        (const float4*)h, srcIdx, dstIdx, dst, N_EDGES, F4);
  };

  // ---- conv1: K padded 11 -> 12 (input cols AND weight rows)
  pad12_kernel<<<cdiv(N_NODES * 12, 256), 256, 0, stream>>>(x, B0, N_NODES * 12);
  edge_scatter_pad<<<cdiv(N_EDGES * 11, 256), 256, 0, stream>>>(
      x, srcIdx, dstIdx, B0, N_EDGES, 11, 12);
  // pad l1w [11,128] -> [12,128]: rows 0..10 copied, row 11 zero
  zero_f32_kernel<<<cdiv(12 * 128, 256), 256, 0, stream>>>(W0p, 12 * 128);
  copy_vec4_kernel<<<cdiv(11 * 128 / 4, 256), 256, 0, stream>>>(
      (float4*)W0p, (const float4*)Wm[0], 11 * 128 / 4);
  gemm(B0, W0p, Bm[0], B1, N_NODES, 12, 128);
  bn_relu(B1, Gm[0], BEm[0], N_NODES, 128);
  gemm(B1, Wm[1], Bm[1], B0, N_NODES, 128, 128);
  bn_relu(B0, Gm[1], BEm[1], N_NODES, 128);            // h2 in B0

  // ---- conv2
  gin_agg(B0, B1, 128);
  gemm(B1, Wm[2], Bm[2], B0, N_NODES, 128, 256);
  bn_relu(B0, Gm[2], BEm[2], N_NODES, 256);
  gemm(B0, Wm[3], Bm[3], B1, N_NODES, 256, 256);
  bn_relu(B1, Gm[3], BEm[3], N_NODES, 256);            // h4 in B1

  // ---- conv3
  gin_agg(B1, B0, 256);
  gemm(B0, Wm[4], Bm[4], B1, N_NODES, 256, 256);
  bn_relu(B1, Gm[4], BEm[4], N_NODES, 256);
  gemm(B1, Wm[5], Bm[5], B0, N_NODES, 256, 256);
  bn_relu(B0, Gm[5], BEm[5], N_NODES, 256);            // h6 in B0

  // ---- global add pool into padded [1008,256]
  zero_f32_kernel<<<cdiv(G_PAD * 256, 256), 256, 0, stream>>>(hg, G_PAD * 256);
  pool_scatter_vec4<<<cdiv(N_NODES * 64, 256), 256, 0, stream>>>(
      (const float4*)B0, batch, hg, N_NODES, 64);

  // ---- head
  gemm(hg, fw1, fb1, hgl, G_PAD, 256, 128);
  bn_relu(hgl, fg, fbe, N_GRAPHS, 128);
  final_linear_kernel<<<N_GRAPHS, 128, 0, stream>>>(hgl, fw2, fb2, out, 128);
}